// HierarchicalPooling_4389456576639
// MI455X (gfx1250) — compile-verified
//
#include <hip/hip_runtime.h>

#define B_DIM   32
#define L_DIM   2048
#define D_DIM   1024
#define KWIN    8
#define NWIN    (L_DIM - KWIN + 1)              /* 2041 windows   */
#define CHUNK_W 128                             /* windows/chunk  */
#define NCHUNK  ((NWIN + CHUNK_W - 1) / CHUNK_W)/* 16 chunks      */
#define TPB     256                             /* 4 channels per thread: 256*4 = 1024 = D */
#define PF_ROWS 16                              /* prefetch distance in l-rows (64 KB)     */

typedef float v4f __attribute__((ext_vector_type(4)));

// Order-preserving map float -> u32 so integer atomicMax implements float max.
__device__ __forceinline__ unsigned int f32_to_ordered_u32(float f) {
    unsigned int u = __float_as_uint(f);
    return (u & 0x80000000u) ? ~u : (u | 0x80000000u);
}
__device__ __forceinline__ float ordered_u32_to_f32(unsigned int u) {
    unsigned int v = (u & 0x80000000u) ? (u ^ 0x80000000u) : ~u;
    return __uint_as_float(v);
}

__global__ void hp_init_kernel(unsigned int* __restrict__ out) {
    int i = blockIdx.x * blockDim.x + threadIdx.x;
    if (i < B_DIM * D_DIM) out[i] = f32_to_ordered_u32(-__builtin_inff());
}

// Window-sum + max over one chunk of windows starting at w0.
// GUARDED=false: all CHUNK_W windows valid, no clamps/predicates in the hot loop
//                -> one base pointer, constant immediate offsets, minimal SALU.
// GUARDED=true : clamp incoming row to L-1, mask max updates for w >= NWIN.
template <bool GUARDED>
__device__ __forceinline__ v4f pool_chunk(const v4f* __restrict__ base_b, int w0) {
    const int strideF4 = D_DIM / 4;            // float4 stride per l row (4 KB)

    v4f ring[KWIN];
    v4f sum = (v4f){0.f, 0.f, 0.f, 0.f};

    // Preload first K-1 rows (always in-bounds: w0+6 <= 1926).
    const v4f* p0 = base_b + (size_t)w0 * strideF4;
#pragma unroll
    for (int i = 0; i < KWIN - 1; ++i) {
        v4f v = __builtin_nontemporal_load(p0 + i * strideF4);
        ring[i] = v;
        sum += v;
    }

    v4f m = (v4f){-__builtin_inff(), -__builtin_inff(),
                  -__builtin_inff(), -__builtin_inff()};

    for (int j = 0; j < CHUNK_W / KWIN; ++j) {
        // Base pointer for this unrolled group; fast path uses immediate offsets.
        const v4f* pj = base_b + (size_t)(w0 + j * KWIN + (KWIN - 1)) * strideF4;
#pragma unroll
        for (int u = 0; u < KWIN; ++u) {
            const int jw = j * KWIN + u;               // window index within chunk
            const v4f* pl;
            if (GUARDED) {
                int l = w0 + jw + (KWIN - 1);
                if (l > L_DIM - 1) l = L_DIM - 1;      // clamp (tail chunk only)
                pl = base_b + (size_t)l * strideF4;
            } else {
                pl = pj + u * strideF4;                // constant immediate offset
            }

            // gfx1250 speculative prefetch of a future row (global_prefetch_b8).
            // Fast path: max prefetch l = w0+127+7+16 <= 1942 < L, in-bounds.
            if (GUARDED) {
                int lpf = w0 + jw + (KWIN - 1) + PF_ROWS;
                if (lpf > L_DIM - 1) lpf = L_DIM - 1;
                __builtin_prefetch((const void*)(base_b + (size_t)lpf * strideF4), 0, 0);
            } else {
                __builtin_prefetch((const void*)(pl + PF_ROWS * strideF4), 0, 0);
            }

            v4f v = __builtin_nontemporal_load(pl);    // global_load_b128 th:NT
            sum += v;

            bool valid = !GUARDED || (w0 + jw < NWIN); // uniform; folds away in fast path
            if (valid) {
                m[0] = fmaxf(m[0], sum[0]);
                m[1] = fmaxf(m[1], sum[1]);
                m[2] = fmaxf(m[2], sum[2]);
                m[3] = fmaxf(m[3], sum[3]);
            }
            sum -= ring[u];                            // row l = w0 + jw leaves window
            ring[(u + KWIN - 1) & (KWIN - 1)] = v;     // incoming row -> its slot
        }
    }
    return m;
}

__global__ __launch_bounds__(TPB) void hp_pool_kernel(const float* __restrict__ x,
                                                      unsigned int* __restrict__ out) {
    const int chunk = blockIdx.x;          // which window-chunk along L
    const int b     = blockIdx.y;          // batch
    const int tid   = threadIdx.x;         // one float4 (4 channels) per thread
    const int w0    = chunk * CHUNK_W;

    const int strideF4 = D_DIM / 4;
    const v4f* base_b  = (const v4f*)x + (size_t)b * L_DIM * strideF4 + tid;

    v4f m;
    if (w0 + CHUNK_W <= NWIN) {
        m = pool_chunk<false>(base_b, w0);   // chunks 0..14: branch-free hot loop
    } else {
        m = pool_chunk<true>(base_b, w0);    // chunk 15: clamped + masked tail
    }

    // max(mean) = max(sum) * (1/8): exact power-of-two scale, applied once.
    unsigned int* o = out + (size_t)b * D_DIM + 4 * tid;
    atomicMax(o + 0, f32_to_ordered_u32(m[0] * 0.125f));
    atomicMax(o + 1, f32_to_ordered_u32(m[1] * 0.125f));
    atomicMax(o + 2, f32_to_ordered_u32(m[2] * 0.125f));
    atomicMax(o + 3, f32_to_ordered_u32(m[3] * 0.125f));
}

__global__ void hp_decode_kernel(unsigned int* __restrict__ out) {
    int i = blockIdx.x * blockDim.x + threadIdx.x;
    if (i < B_DIM * D_DIM) {
        float f = ordered_u32_to_f32(out[i]);
        ((float*)out)[i] = f;   // in-place: each thread owns one element
    }
}

extern "C" void kernel_launch(void* const* d_in, const int* in_sizes, int n_in,
                              void* d_out, int out_size, void* d_ws, size_t ws_size,
                              hipStream_t stream) {
    (void)in_sizes; (void)n_in; (void)d_ws; (void)ws_size; (void)out_size;
    const float* x = (const float*)d_in[0];
    unsigned int* out_u = (unsigned int*)d_out;

    hp_init_kernel<<<(B_DIM * D_DIM + 255) / 256, 256, 0, stream>>>(out_u);

    dim3 grid(NCHUNK, B_DIM);
    hp_pool_kernel<<<grid, TPB, 0, stream>>>(x, out_u);

    hp_decode_kernel<<<(B_DIM * D_DIM + 255) / 256, 256, 0, stream>>>(out_u);
}